// memristor_dense_31756988187014
// MI455X (gfx1250) — compile-verified
//
#include <hip/hip_runtime.h>
#include <stdint.h>

// Problem constants (fixed by the reference setup_inputs()).
#define N_IN    1024
#define N_I     1025          // n_in + 1 bias row
#define N_OUTP  512           // output pairs (n_out)
#define BATCH   128
#define V_REF_C 0.25f
#define G_MIN_C 0.1f
#define G_MAX_C 1.0f
#define K_V_C   0.5f

// Main-kernel tiling.
#define JPB           8       // output pairs per block (16 interleaved columns)
#define SUBI          32      // i-rows per pipeline stage
#define NSUB_TOTAL    33      // ceil(1025/32)
#define CHUNKS        11      // i-chunks (grid.y); 11*3 = 33 sub-chunks
#define SUB_PER_CHUNK 3
#define XPITCH        36      // padded floats per batch-row of x tile (16B aligned, few bank conflicts)

__device__ __forceinline__ uint32_t lds_off_of(const void* p) {
  // Generic pointers to LDS carry the byte offset in the low 32 bits.
  return (uint32_t)(uintptr_t)p;
}

// Direct v_exp_f32 / v_log_f32 (log2) — avoids OCML's denorm-fixup wrapper so
// the trans pipe, not the VALU pipe, is the bottleneck. ~1 ulp; output-denorm
// flush only affects terms < 2^-126, negligible in the 1025-term sums.
__device__ __forceinline__ float fast_exp2(float a) {
#if __has_builtin(__builtin_amdgcn_exp2f)
  return __builtin_amdgcn_exp2f(a);
#else
  return exp2f(a);
#endif
}
__device__ __forceinline__ float fast_log2(float a) {
#if __has_builtin(__builtin_amdgcn_logf)
  return __builtin_amdgcn_logf(a);   // v_log_f32 computes log2(x)
#else
  return log2f(a);
#endif
}

// ---------------------------------------------------------------------------
// Kernel 1: max|w| over w_pos, w_neg, b_pos, b_neg  (bitwise atomicMax on
// non-negative floats is order-preserving -> deterministic result).
// ---------------------------------------------------------------------------
__global__ __launch_bounds__(256) void memristor_maxw_kernel(
    const float* __restrict__ wp, const float* __restrict__ wn,
    const float* __restrict__ bp, const float* __restrict__ bn,
    unsigned* __restrict__ maxbits) {
  const int tid = blockIdx.x * blockDim.x + threadIdx.x;
  const int stride = gridDim.x * blockDim.x;
  float m = 0.0f;
  for (int idx = tid; idx < N_IN * N_OUTP; idx += stride) {
    m = fmaxf(m, fabsf(wp[idx]));
    m = fmaxf(m, fabsf(wn[idx]));
  }
  for (int idx = tid; idx < N_OUTP; idx += stride) {
    m = fmaxf(m, fabsf(bp[idx]));
    m = fmaxf(m, fabsf(bn[idx]));
  }
  __shared__ float red[256];
  red[threadIdx.x] = m;
  __syncthreads();
  for (int s = 128; s > 0; s >>= 1) {
    if (threadIdx.x < s) red[threadIdx.x] = fmaxf(red[threadIdx.x], red[threadIdx.x + s]);
    __syncthreads();
  }
  if (threadIdx.x == 0) atomicMax(maxbits, __float_as_uint(red[0]));
}

// ---------------------------------------------------------------------------
// Kernel 2: fold interleave/bias/k_G/V_REF/log2(n_param)+1 into
// EG[i][jp] = {E_pos, E_neg, GV_pos, GV_neg}.
// ---------------------------------------------------------------------------
__global__ __launch_bounds__(256) void memristor_prep_kernel(
    const float* __restrict__ wp, const float* __restrict__ wn,
    const float* __restrict__ bp, const float* __restrict__ bn,
    const float* __restrict__ npar, const unsigned* __restrict__ maxbits,
    float4* __restrict__ EG) {
  const int idx = blockIdx.x * blockDim.x + threadIdx.x;
  if (idx >= N_I * N_OUTP) return;
  const int i = idx / N_OUTP;
  const int jp = idx - i * N_OUTP;
  const float max_w = __uint_as_float(*maxbits);
  const float k_G = (G_MAX_C - G_MIN_C) / max_w;
  const float wpv = (i < N_IN) ? wp[i * N_OUTP + jp] : bp[jp];
  const float wnv = (i < N_IN) ? wn[i * N_OUTP + jp] : bn[jp];
  const float Gp = (fabsf(k_G * wpv) + G_MIN_C) * V_REF_C;
  const float Gn = (fabsf(k_G * wnv) + G_MIN_C) * V_REF_C;
  const float Ep = fast_log2(npar[i * (2 * N_OUTP) + 2 * jp + 0]) + 1.0f;
  const float En = fast_log2(npar[i * (2 * N_OUTP) + 2 * jp + 1]) + 1.0f;
  EG[idx] = make_float4(Ep, En, Gp, Gn);
}

// ---------------------------------------------------------------------------
// Kernel 3: main fused contraction. CDNA5 async global->LDS double buffering.
// Each block: all 128 batches x 8 output pairs, one i-chunk (3 x 32 rows).
// ---------------------------------------------------------------------------
#define ISSUE_STAGE(buf, i0base)                                               \
  {                                                                            \
    const uint32_t xl0 = lds_off_of(&xtile[(buf)][0]);                         \
    const uint32_t el0 = lds_off_of(&egtile[(buf)][0]);                        \
    for (int q = 0; q < 4; ++q) {                                              \
      const int l = q * 256 + t;                                               \
      const int bb = l >> 3;                                                   \
      const int ff = l & 7;                                                    \
      int col = (i0base) + ff * 4;                                             \
      col = col > (N_IN - 4) ? (N_IN - 4) : col;  /* clamp for bias stage */   \
      const uint32_t go = (uint32_t)((bb * N_IN + col) * 4);                   \
      const uint32_t lo = xl0 + (uint32_t)((bb * XPITCH + ff * 4) * 4);        \
      asm volatile("global_load_async_to_lds_b128 %0, %1, %2"                  \
                   :: "v"(lo), "v"(go), "s"(xbase) : "memory");                \
    }                                                                          \
    const int isub = t >> 3;                                                   \
    const int cc = t & 7;                                                      \
    int row = (i0base) + isub;                                                 \
    row = row > (N_I - 1) ? (N_I - 1) : row;                                   \
    const uint32_t go2 = (uint32_t)(((uint32_t)row * N_OUTP + jp0 + cc) * 16u);\
    const uint32_t lo2 = el0 + (uint32_t)(t * 16);                             \
    asm volatile("global_load_async_to_lds_b128 %0, %1, %2"                    \
                 :: "v"(lo2), "v"(go2), "s"(egbase) : "memory");               \
  }

__global__ __launch_bounds__(256) void memristor_main_kernel(
    const float* __restrict__ x, const float4* __restrict__ EG,
    float* __restrict__ part) {
  __shared__ float  xtile[2][BATCH * XPITCH];   // 36864 B
  __shared__ float4 egtile[2][SUBI * JPB];      //  8192 B

  const int t = (int)threadIdx.x;
  const int b = t & (BATCH - 1);   // batch owned by this thread
  const int g = t >> 7;            // 0/1 -> which group of 4 output pairs
  const int jp0 = (int)blockIdx.x * JPB;
  const int s0 = (int)blockIdx.y * SUB_PER_CHUNK;
  const int s1 = s0 + SUB_PER_CHUNK;

  const uint64_t xbase = (uint64_t)(uintptr_t)x;
  const uint64_t egbase = (uint64_t)(uintptr_t)EG;

  float acc0 = 0.0f, acc1 = 0.0f, acc2 = 0.0f, acc3 = 0.0f;

  ISSUE_STAGE(0, s0 * SUBI);

  for (int s = s0; s < s1; ++s) {
    const int cur = (s - s0) & 1;
    const int i0 = s * SUBI;
    if (s + 1 < s1) {
      ISSUE_STAGE(cur ^ 1, (s + 1) * SUBI);
      asm volatile("s_wait_asynccnt 0x5" ::: "memory");  // current stage (5 ops/wave) done
    } else {
      asm volatile("s_wait_asynccnt 0x0" ::: "memory");
    }
    __syncthreads();  // all waves' DMA landed

    const float*  xb = &xtile[cur][b * XPITCH];
    const float4* eb = &egtile[cur][g * 4];
    const int ni = (N_I - i0) < SUBI ? (N_I - i0) : SUBI;

    for (int ii = 0; ii < ni; ++ii) {
      const int gi = i0 + ii;
      float L;
      if (gi == N_IN) {
        L = 1.0f;                                   // bias row: vr = 2 -> log2 = 1
      } else {
        float xv = fminf(fmaxf(xb[ii], 0.0f), 1.0f);
        L = (xv > 0.0f) ? fast_log2(2.0f * xv) : -1.0e30f;  // vr==0 -> exp2 -> 0
      }
      const float4* er = eb + ii * JPB;             // wave-uniform LDS broadcasts
      const float4 e0 = er[0];
      const float4 e1 = er[1];
      const float4 e2 = er[2];
      const float4 e3 = er[3];
      acc0 += e0.z * fast_exp2(e0.x * L) - e0.w * fast_exp2(e0.y * L);
      acc1 += e1.z * fast_exp2(e1.x * L) - e1.w * fast_exp2(e1.y * L);
      acc2 += e2.z * fast_exp2(e2.x * L) - e2.w * fast_exp2(e2.y * L);
      acc3 += e3.z * fast_exp2(e3.x * L) - e3.w * fast_exp2(e3.y * L);
    }
    __syncthreads();  // release 'cur' buffer before next prefetch overwrites it
  }

  float* prow = part + ((size_t)blockIdx.y * BATCH + b) * N_OUTP + jp0 + g * 4;
  prow[0] = acc0;
  prow[1] = acc1;
  prow[2] = acc2;
  prow[3] = acc3;
}

// ---------------------------------------------------------------------------
// Kernel 4: deterministic fixed-order combine of the CHUNKS partials + scale.
// y = (I_pos - I_neg) / (K_V * k_G) = sum * max_w / (K_V * (G_MAX - G_MIN))
// ---------------------------------------------------------------------------
__global__ __launch_bounds__(256) void memristor_finalize_kernel(
    const float* __restrict__ part, const unsigned* __restrict__ maxbits,
    float* __restrict__ out) {
  const int idx = blockIdx.x * blockDim.x + threadIdx.x;  // over 128*512
  if (idx >= BATCH * N_OUTP) return;
  float s = 0.0f;
  for (int c = 0; c < CHUNKS; ++c) s += part[(size_t)c * BATCH * N_OUTP + idx];
  const float max_w = __uint_as_float(*maxbits);
  out[idx] = s * (max_w / (K_V_C * (G_MAX_C - G_MIN_C)));
}

// ---------------------------------------------------------------------------
extern "C" void kernel_launch(void* const* d_in, const int* in_sizes, int n_in,
                              void* d_out, int out_size, void* d_ws, size_t ws_size,
                              hipStream_t stream) {
  (void)in_sizes; (void)n_in; (void)out_size; (void)ws_size;
  const float* x  = (const float*)d_in[0];
  const float* wp = (const float*)d_in[1];
  const float* wn = (const float*)d_in[2];
  const float* bp = (const float*)d_in[3];
  const float* bn = (const float*)d_in[4];
  const float* np = (const float*)d_in[5];

  unsigned* maxbits = (unsigned*)d_ws;
  float4*   EG      = (float4*)((char*)d_ws + 256);
  float*    part    = (float*)((char*)d_ws + 256 + (size_t)N_I * N_OUTP * sizeof(float4));
  float*    out     = (float*)d_out;

  hipMemsetAsync(maxbits, 0, sizeof(unsigned), stream);

  memristor_maxw_kernel<<<256, 256, 0, stream>>>(wp, wn, bp, bn, maxbits);

  memristor_prep_kernel<<<(N_I * N_OUTP + 255) / 256, 256, 0, stream>>>(
      wp, wn, bp, bn, np, maxbits, EG);

  dim3 grid(N_OUTP / JPB, CHUNKS);  // 64 x 11
  memristor_main_kernel<<<grid, 256, 0, stream>>>(x, EG, part);

  memristor_finalize_kernel<<<(BATCH * N_OUTP + 255) / 256, 256, 0, stream>>>(
      part, maxbits, out);
}